// Model_33569464386210
// MI455X (gfx1250) — compile-verified
//
#include <hip/hip_runtime.h>
#include <math.h>

typedef __attribute__((ext_vector_type(2))) float v2f;
typedef __attribute__((ext_vector_type(8))) float v8f;

#define NM_ 200000
#define NC_ 50000
#define DD_ 128
#define HH_ 64
#define EE_ 1000000
#define EL_ 100000

// ---------------- atomic float max via int trick (buffer init to -inf) -----
__device__ __forceinline__ void atomicMaxF(float* addr, float val) {
    if (val >= 0.0f) {
        atomicMax((int*)addr, __float_as_int(val));
    } else {
        atomicMin((unsigned int*)addr, (unsigned int)__float_as_int(val));
    }
}

// ---------------- C[M,64] = A[M,K] @ B[K,64], f32 WMMA 16x16x4 -------------
// one wave per 16x16 tile; 4 waves/block cover the 64 columns.
template <int K>
__global__ void gemm_wmma(const float* __restrict__ A, const float* __restrict__ B,
                          float* __restrict__ C, int M) {
    const int lane = threadIdx.x & 31;
    const int wave = threadIdx.x >> 5;       // 0..3 -> column tile
    const int row0 = blockIdx.x * 16;
    const int n    = lane & 15;
    const int hi   = lane >> 4;              // 0: K pair {0,1}, 1: K pair {2,3}
    const int col  = wave * 16 + n;
    const float* arow = A + (size_t)(row0 + n) * K;
    v8f acc = {};
#pragma unroll
    for (int k0 = 0; k0 < K; k0 += 4) {
        const int ka = k0 + 2 * hi;
        v2f a, b;
        a.x = arow[ka];
        a.y = arow[ka + 1];
        b.x = B[ka * 64 + col];
        b.y = B[(ka + 1) * 64 + col];
        acc = __builtin_amdgcn_wmma_f32_16x16x4_f32(false, a, false, b,
                                                    (short)0, acc, false, false);
    }
    float* crow = C + ((size_t)(row0 + 8 * hi)) * 64 + col;
#pragma unroll
    for (int r = 0; r < 8; ++r) crow[(size_t)r * 64] = acc[r];
}

// ---- decoder GEMM: T[e,:] = concat(ZM2[row[e]], ZC2[col[e]]) @ Wd1 + bd1 --
__global__ void gemm_decoder(const float* __restrict__ ZM2, const float* __restrict__ ZC2,
                             const int* __restrict__ rowI, const int* __restrict__ colI,
                             const float* __restrict__ Wd1, const float* __restrict__ bd1,
                             float* __restrict__ T) {
    const int lane = threadIdx.x & 31;
    const int wave = threadIdx.x >> 5;
    const int row0 = blockIdx.x * 16;
    const int n    = lane & 15;
    const int hi   = lane >> 4;
    const int col  = wave * 16 + n;
    const int ri = rowI[row0 + n];
    const int ci = colI[row0 + n];
    const float* pm = ZM2 + (size_t)ri * 64;
    const float* pc = ZC2 + (size_t)ci * 64;
    v8f acc = {};
#pragma unroll
    for (int k0 = 0; k0 < 128; k0 += 4) {
        const int ka = k0 + 2 * hi;          // even, so ka and ka+1 same half
        v2f a, b;
        if (ka < 64) { a.x = pm[ka];      a.y = pm[ka + 1];      }
        else         { a.x = pc[ka - 64]; a.y = pc[ka - 63];     }
        b.x = Wd1[ka * 64 + col];
        b.y = Wd1[(ka + 1) * 64 + col];
        acc = __builtin_amdgcn_wmma_f32_16x16x4_f32(false, a, false, b,
                                                    (short)0, acc, false, false);
    }
    const float bv = bd1[col];
    float* crow = T + ((size_t)(row0 + 8 * hi)) * 64 + col;
#pragma unroll
    for (int r = 0; r < 8; ++r) crow[(size_t)r * 64] = acc[r] + bv;
}

// ---------------- wa[k] = sum_j W[k,j] * a[j]  (W: [K,64]) ------------------
__global__ void compute_wa(const float* __restrict__ W, const float* __restrict__ a,
                           float* __restrict__ wa, int K) {
    int k = threadIdx.x;
    if (k >= K) return;
    float s = 0.0f;
#pragma unroll
    for (int j = 0; j < 64; ++j) s += W[k * 64 + j] * a[j];
    wa[k] = s;
}

// ---------------- al[i] = dot(X[i,:], w)  (one wave per row) ----------------
__global__ void matvec(const float* __restrict__ X, const float* __restrict__ w,
                       float* __restrict__ out, int rows, int K) {
    int wid  = (int)((blockIdx.x * (size_t)blockDim.x + threadIdx.x) >> 5);
    int lane = threadIdx.x & 31;
    if (wid >= rows) return;
    const float* x = X + (size_t)wid * K;
    float s = 0.0f;
    for (int k = lane; k < K; k += 32) s += x[k] * w[k];
#pragma unroll
    for (int off = 16; off; off >>= 1) s += __shfl_xor(s, off, 32);
    if (lane == 0) out[wid] = s;
}

__global__ void fillk(float* __restrict__ p, float v, size_t n) {
    size_t i = blockIdx.x * (size_t)blockDim.x + threadIdx.x;
    if (i < n) p[i] = v;
}

// -------- P1: alpha = leaky_relu(als[src]+ald[dst]); segment max -----------
__global__ void edge_p1(const float* __restrict__ ALS, const float* __restrict__ ALD,
                        const int* __restrict__ src, const int* __restrict__ dst,
                        float* __restrict__ ALPHA, float* __restrict__ MBUF) {
    int e = blockIdx.x * blockDim.x + threadIdx.x;
    if (e >= EE_) return;
    float a = ALS[src[e]] + ALD[dst[e]];
    a = (a > 0.0f) ? a : 0.2f * a;
    ALPHA[e] = a;
    atomicMaxF(&MBUF[dst[e]], a);
}

// -------- P2: e = exp(alpha - m[dst]); denom += e (in-place in ALPHA) ------
__global__ void edge_p2(float* __restrict__ ALPHA, const float* __restrict__ MBUF,
                        const int* __restrict__ dst, float* __restrict__ DEN) {
    int e = blockIdx.x * blockDim.x + threadIdx.x;
    if (e >= EE_) return;
    int d = dst[e];
    float ee = __expf(ALPHA[e] - MBUF[d]);
    ALPHA[e] = ee;
    atomicAdd(&DEN[d], ee);
}

// -------- P3: OUT[dst] += (e/denom) * HS[src]  (thread = edge x 4 cols) ----
__global__ void edge_p3(const float* __restrict__ ALPHA, const float* __restrict__ DEN,
                        const int* __restrict__ src, const int* __restrict__ dst,
                        const float* __restrict__ HS, float* __restrict__ OUT) {
    int t = blockIdx.x * blockDim.x + threadIdx.x;
    int e = t >> 4;
    if (e >= EE_) return;
    int q = (t & 15) * 4;
    int s = src[e], d = dst[e];
    float w = ALPHA[e] / fmaxf(DEN[d], 1e-16f);
    const float4 v = *(const float4*)(HS + (size_t)s * 64 + q);
    float* o = OUT + (size_t)d * 64 + q;
    atomicAdd(o + 0, w * v.x);
    atomicAdd(o + 1, w * v.y);
    atomicAdd(o + 2, w * v.z);
    atomicAdd(o + 3, w * v.w);
}

// -------- bias (+ optional relu) over [Nd,64] ------------------------------
__global__ void bias_relu(float* __restrict__ OUT, const float* __restrict__ b,
                          int relu, size_t n) {
    size_t i = blockIdx.x * (size_t)blockDim.x + threadIdx.x;
    if (i >= n) return;
    float v = OUT[i] + b[i & 63];
    OUT[i] = relu ? fmaxf(v, 0.0f) : v;
}

// -------- final decoder: out[e,:2] = T[e,:] @ Wd2 + bd2 --------------------
__global__ void decoder_out(const float* __restrict__ T, const float* __restrict__ Wd2,
                            const float* __restrict__ bd2, float* __restrict__ out) {
    int e = blockIdx.x * blockDim.x + threadIdx.x;
    if (e >= EL_) return;
    const float* t = T + (size_t)e * 64;
    float s0 = bd2[0], s1 = bd2[1];
#pragma unroll
    for (int h = 0; h < 64; ++h) {
        float tv = t[h];
        s0 += tv * Wd2[h * 2 + 0];
        s1 += tv * Wd2[h * 2 + 1];
    }
    out[2 * e + 0] = s0;
    out[2 * e + 1] = s1;
}

// ---------------------------------------------------------------------------
struct GatWS {
    float *HS, *ALS, *ALD, *ALPHA, *MBUF, *DEN, *WAS, *WAD;
};

static void run_gat(const float* Hsrc, int Ns, int Din,
                    const float* Hdst, int Nd,
                    const float* Ws, const float* Wd,
                    const float* as_, const float* ad_, const float* bias,
                    const int* src, const int* dst,
                    float* OUT, int relu, const GatWS& w, hipStream_t s) {
    if (Din == 128) gemm_wmma<128><<<Ns / 16, 128, 0, s>>>(Hsrc, Ws, w.HS, Ns);
    else            gemm_wmma<64> <<<Ns / 16, 128, 0, s>>>(Hsrc, Ws, w.HS, Ns);
    compute_wa<<<1, Din, 0, s>>>(Ws, as_, w.WAS, Din);
    compute_wa<<<1, Din, 0, s>>>(Wd, ad_, w.WAD, Din);
    matvec<<<Ns / 4, 128, 0, s>>>(Hsrc, w.WAS, w.ALS, Ns, Din);
    matvec<<<Nd / 4, 128, 0, s>>>(Hdst, w.WAD, w.ALD, Nd, Din);
    fillk<<<(Nd + 255) / 256, 256, 0, s>>>(w.MBUF, -INFINITY, (size_t)Nd);
    fillk<<<(Nd + 255) / 256, 256, 0, s>>>(w.DEN, 0.0f, (size_t)Nd);
    fillk<<<((size_t)Nd * 64 + 255) / 256, 256, 0, s>>>(OUT, 0.0f, (size_t)Nd * 64);
    edge_p1<<<EE_ / 256, 256, 0, s>>>(w.ALS, w.ALD, src, dst, w.ALPHA, w.MBUF);
    edge_p2<<<EE_ / 256, 256, 0, s>>>(w.ALPHA, w.MBUF, dst, w.DEN);
    edge_p3<<<(EE_ * 16) / 256, 256, 0, s>>>(w.ALPHA, w.DEN, src, dst, w.HS, OUT);
    bias_relu<<<((size_t)Nd * 64) / 256, 256, 0, s>>>(OUT, bias, relu, (size_t)Nd * 64);
}

extern "C" void kernel_launch(void* const* d_in, const int* in_sizes, int n_in,
                              void* d_out, int out_size, void* d_ws, size_t ws_size,
                              hipStream_t stream) {
    const float* x_m    = (const float*)d_in[0];
    const float* x_c    = (const float*)d_in[1];
    const int*   src_mc = (const int*)d_in[2];
    const int*   dst_mc = (const int*)d_in[3];
    const int*   src_cm = (const int*)d_in[4];
    const int*   dst_cm = (const int*)d_in[5];
    const int*   rowI   = (const int*)d_in[6];
    const int*   colI   = (const int*)d_in[7];
    const float *W1s_mc = (const float*)d_in[8],  *W1d_mc = (const float*)d_in[9];
    const float *a1s_mc = (const float*)d_in[10], *a1d_mc = (const float*)d_in[11], *b1_mc = (const float*)d_in[12];
    const float *W1s_cm = (const float*)d_in[13], *W1d_cm = (const float*)d_in[14];
    const float *a1s_cm = (const float*)d_in[15], *a1d_cm = (const float*)d_in[16], *b1_cm = (const float*)d_in[17];
    const float *W2s_mc = (const float*)d_in[18], *W2d_mc = (const float*)d_in[19];
    const float *a2s_mc = (const float*)d_in[20], *a2d_mc = (const float*)d_in[21], *b2_mc = (const float*)d_in[22];
    const float *W2s_cm = (const float*)d_in[23], *W2d_cm = (const float*)d_in[24];
    const float *a2s_cm = (const float*)d_in[25], *a2d_cm = (const float*)d_in[26], *b2_cm = (const float*)d_in[27];
    const float *Wd1 = (const float*)d_in[28], *bd1 = (const float*)d_in[29];
    const float *Wd2 = (const float*)d_in[30], *bd2 = (const float*)d_in[31];

    float* w = (float*)d_ws;
    float* ZM    = w;                 w += (size_t)NM_ * 64;
    float* ZC    = w;                 w += (size_t)NC_ * 64;
    float* ZM2   = w;                 w += (size_t)NM_ * 64;
    float* ZC2   = w;                 w += (size_t)NC_ * 64;
    GatWS g;
    g.HS    = w;                      w += (size_t)NM_ * 64;   // also reused as decoder T
    g.ALS   = w;                      w += NM_;
    g.ALD   = w;                      w += NM_;
    g.ALPHA = w;                      w += EE_;
    g.MBUF  = w;                      w += NM_;
    g.DEN   = w;                      w += NM_;
    g.WAS   = w;                      w += 128;
    g.WAD   = w;                      w += 128;

    // Layer 1: z_c = relu(gat(x_m -> x_c)), z_m = relu(gat(x_c -> x_m))
    run_gat(x_m, NM_, 128, x_c, NC_, W1s_mc, W1d_mc, a1s_mc, a1d_mc, b1_mc,
            src_mc, dst_mc, ZC, 1, g, stream);
    run_gat(x_c, NC_, 128, x_m, NM_, W1s_cm, W1d_cm, a1s_cm, a1d_cm, b1_cm,
            src_cm, dst_cm, ZM, 1, g, stream);
    // Layer 2 (no relu)
    run_gat(ZM, NM_, 64, ZC, NC_, W2s_mc, W2d_mc, a2s_mc, a2d_mc, b2_mc,
            src_mc, dst_mc, ZC2, 0, g, stream);
    run_gat(ZC, NC_, 64, ZM, NM_, W2s_cm, W2d_cm, a2s_cm, a2d_cm, b2_cm,
            src_cm, dst_cm, ZM2, 0, g, stream);

    // Decoder: fused gather-concat GEMM + tiny final layer
    float* T = g.HS;  // reuse scratch
    gemm_decoder<<<EL_ / 16, 128, 0, stream>>>(ZM2, ZC2, rowI, colI, Wd1, bd1, T);
    decoder_out<<<(EL_ + 127) / 128, 128, 0, stream>>>(T, Wd2, bd2, (float*)d_out);
}